// CityModel_64965675319620
// MI455X (gfx1250) — compile-verified
//
#include <hip/hip_runtime.h>
#include <hip/hip_bf16.h>
#include <math.h>

// ---------------- model dims (from reference) ----------------
namespace {
constexpr int Bn   = 16;
constexpr int Sn   = 256;
constexpr int En   = 2048;
constexpr int Tn   = 48;
constexpr int NGn  = Bn * 24;      // 384 graphs
constexpr int Nn   = NGn * Sn;     // 98304 nodes
constexpr int NEn  = NGn * En;     // 786432 edges
constexpr int BSn  = Bn * Sn;      // 4096 sequences
constexpr int NWAVES_LSTM = 4;     // 128-thread blocks for LSTM kernels
}

typedef __attribute__((ext_vector_type(2))) float v2f;
typedef __attribute__((ext_vector_type(8))) float v8f;

// fp32 WMMA: D(16x16,f32) = A(16x4,f32) x B(4x16,f32) + C
//   A: lane L<16 -> (M=L, K=k0+{0,1}); lane L>=16 -> (M=L-16, K=k0+{2,3})
//   B: lane L<16 -> (K=k0+{0,1}, N=L); lane L>=16 -> (K=k0+{2,3}, N=L-16)
//   C/D: vgpr r, lane L -> (M = r + 8*(L>=16), N = L%16)
__device__ __forceinline__ v8f wmma_f32(v2f a, v2f b, v8f c) {
  return __builtin_amdgcn_wmma_f32_16x16x4_f32(false, a, false, b, (short)0, c, false, false);
}

__device__ __forceinline__ v2f ld2(const float* p) { return *(const v2f*)p; }  // 8B-aligned pair

__device__ __forceinline__ float sigf(float x) { return 1.0f / (1.0f + expf(-x)); }
__device__ __forceinline__ float reluf(float x) { return fmaxf(x, 0.0f); }

// ---- CDNA5 async global->LDS copies (ASYNCcnt-tracked) ----
// Generic pointers to __shared__ carry the LDS offset in the low 32 bits (aperture layout),
// which is the VDST (LDS address) operand of GLOBAL_LOAD_ASYNC_TO_LDS_*.
__device__ __forceinline__ void async_b128(const void* g, void* l) {
  const unsigned long long ga = (unsigned long long)g;
  const unsigned la = (unsigned)(unsigned long long)l;
  asm volatile("global_load_async_to_lds_b128 %0, %1, off" :: "v"(la), "v"(ga) : "memory");
}
__device__ __forceinline__ void async_b64(const void* g, void* l) {
  const unsigned long long ga = (unsigned long long)g;
  const unsigned la = (unsigned)(unsigned long long)l;
  asm volatile("global_load_async_to_lds_b64 %0, %1, off" :: "v"(la), "v"(ga) : "memory");
}
__device__ __forceinline__ void async_wait0() {
  asm volatile("s_wait_asynccnt 0x0" ::: "memory");
}

// ---------------- kernel 1: node feature embed -> x[N,32] ----------------
__global__ void embed_kernel(const float* __restrict__ sta_aqi,  // [B,S,24]
                             const float* __restrict__ sta_poi,  // [B,S,5]
                             const float* __restrict__ Waqi, const float* __restrict__ baqi,
                             const float* __restrict__ Wpoi, const float* __restrict__ bpoi,
                             float* __restrict__ xf)              // [N,32]
{
  const int n = blockIdx.x * blockDim.x + threadIdx.x;
  if (n >= Nn) return;
  const int g = n / Sn, s = n - g * Sn;
  const int b = g / 24, t = g - b * 24;
  const float av = sta_aqi[((size_t)(b * Sn + s)) * 24 + t];
  float p[5];
#pragma unroll
  for (int c = 0; c < 5; ++c) p[c] = sta_poi[(size_t)(b * Sn + s) * 5 + c];
#pragma unroll
  for (int j = 0; j < 16; ++j) xf[(size_t)n * 32 + j] = reluf(av * Waqi[j] + baqi[j]);
#pragma unroll
  for (int j = 0; j < 16; ++j) {
    float acc = bpoi[j];
#pragma unroll
    for (int c = 0; c < 5; ++c) acc += p[c] * Wpoi[c * 16 + j];
    xf[(size_t)n * 32 + 16 + j] = reluf(acc);
  }
}

// ---------------- kernel 2: global features u_base[384,32] ----------------
__global__ void ubase_kernel(const float* __restrict__ city_u,  // [B,24,64]
                             const float* __restrict__ sta_wea, // [B,24,5]
                             const float* __restrict__ Wc, const float* __restrict__ bc,
                             const float* __restrict__ Ww, const float* __restrict__ bw,
                             float* __restrict__ ubase)          // [384,32]
{
  const int i = blockIdx.x * blockDim.x + threadIdx.x;
  if (i >= 384 * 32) return;
  const int row = i >> 5, j = i & 31;
  float v;
  if (j < 16) {
    float acc = bc[j];
    for (int k = 0; k < 64; ++k) acc += city_u[(size_t)row * 64 + k] * Wc[k * 16 + j];
    v = acc;
  } else {
    const int jj = j - 16;
    float acc = bw[jj];
#pragma unroll
    for (int k = 0; k < 5; ++k) acc += sta_wea[(size_t)row * 5 + k] * Ww[k * 16 + jj];
    v = acc;
  }
  ubase[i] = reluf(v);
}

// ---------------- kernel 3: edge MLP + atomic scatter (WMMA, async gather) ----------------
// Weights in LDS are K-pair interleaved: Wl[(k>>1)*128 + 2*j + (k&1)] so each B operand
// is one aligned ds_load_b64 (no re-pairing movs).
__global__ void gnn_msg_kernel(const float* __restrict__ xf,
                               const int* __restrict__ conn,   // [B,E,2]
                               const float* __restrict__ sta_w,// [B,24,E,2]
                               const float* __restrict__ Wn1,  // [66,64]
                               const float* __restrict__ bn1,  // [64]
                               float* __restrict__ sums,       // [N,64]
                               float* __restrict__ cntf)       // [N]
{
  extern __shared__ float smem[];
  float* Wl = smem;                         // 34 k-pairs x 128  (K padded 66->68)
  float* Al = Wl + 68 * 64;                 // 8 waves * [16][68]
  int* colL = (int*)(Al + 8 * 16 * 68);     // 8 waves * [16]

  const int tid = threadIdx.x;
  const int wave = tid >> 5, lane = tid & 31;
  const int m = lane & 15, hi = lane >> 4, koff = hi << 1;

  for (int i = tid; i < 68 * 64; i += blockDim.x) {
    const int k = i >> 6, j = i & 63;
    Wl[((k >> 1) << 7) + (j << 1) + (k & 1)] = (k < 66) ? Wn1[k * 64 + j] : 0.0f;
  }

  float* Aw = Al + wave * (16 * 68);
  int* colw = colL + wave * 16;
  // pad columns 66..67 of every edge row are always zero: write once
  for (int ee = lane; ee < 16; ee += 32) { Aw[ee * 68 + 66] = 0.0f; Aw[ee * 68 + 67] = 0.0f; }

  const int totalWaves = gridDim.x * (blockDim.x >> 5);
  const int waveId = blockIdx.x * (blockDim.x >> 5) + wave;
  const int numTiles = NEn / 16;            // 49152; grid chosen so uniform
  __syncthreads();

  for (int tile = waveId; tile < numTiles; tile += totalWaves) {
    const int eb = tile << 4;
    // stage x[row]/x[col]: 16 edges x 16 b128 chunks, per-lane async gather
    for (int idx = lane; idx < 256; idx += 32) {
      const int ee = idx >> 4, ch = idx & 15;
      const int which = ch >> 3;            // 0 = row part, 1 = col part
      const int co = ch & 7;                // 16B chunk within 32 floats
      const int ek = eb + ee;
      const int g = ek >> 11;               // /E (2048)
      const int e = ek & 2047;
      const int node = conn[((g & (Bn - 1)) * En + e) * 2 + which] + g * Sn;
      async_b128(xf + (size_t)node * 32 + co * 4, Aw + ee * 68 + which * 32 + co * 4);
    }
    if (lane < 16) {
      const int ek = eb + lane;
      const int g = ek >> 11, e = ek & 2047;
      async_b64(sta_w + ((size_t)g * En + e) * 2, Aw + lane * 68 + 64);   // edge_attr
      colw[lane] = conn[((g & (Bn - 1)) * En + e) * 2 + 1] + g * Sn;
    }
    async_wait0();
    __syncthreads();

    // 4 independent accumulator chains per k-step (hides WMMA->WMMA RAW latency)
    v8f acc[4] = {v8f{}, v8f{}, v8f{}, v8f{}};
#pragma unroll
    for (int ks = 0; ks < 17; ++ks) {        // K = 68
      const int k0 = (ks << 2) + koff;
      const v2f a = ld2(Aw + m * 68 + k0);
      const float* wrow = Wl + ((k0 >> 1) << 7);
#pragma unroll
      for (int nt = 0; nt < 4; ++nt)
        acc[nt] = wmma_f32(a, ld2(wrow + ((nt * 16 + m) << 1)), acc[nt]);
    }
#pragma unroll
    for (int nt = 0; nt < 4; ++nt) {
      const float bias = bn1[nt * 16 + m];
#pragma unroll
      for (int r = 0; r < 8; ++r) {
        const int rowi = r + hi * 8;
        const float v = reluf(acc[nt][r] + bias);
        atomicAdd(&sums[(size_t)colw[rowi] * 64 + nt * 16 + m], v);
      }
    }
    if (m == 0) {
#pragma unroll
      for (int r = 0; r < 8; ++r) atomicAdd(&cntf[colw[r + hi * 8]], 1.0f);
    }
    __syncthreads();
  }
}

// ---------------- kernel 4: node MLP (WMMA) -> hxseq[BS,24,64] ----------------
__global__ void gnn_node_kernel(const float* __restrict__ xf, const float* __restrict__ sums,
                                const float* __restrict__ cntf, const float* __restrict__ ubase,
                                const float* __restrict__ Wn2, const float* __restrict__ bn2,
                                float* __restrict__ hxseq)
{
  extern __shared__ float smem[];
  float* Wl = smem;                 // 64 k-pairs x 128 (K=128, interleaved)
  float* Al = Wl + 128 * 64;        // 8 waves * [16][128]
  const int tid = threadIdx.x, wave = tid >> 5, lane = tid & 31;
  const int m = lane & 15, hi = lane >> 4, koff = hi << 1;

  for (int i = tid; i < 128 * 64; i += blockDim.x) {
    const int k = i >> 6, j = i & 63;
    Wl[((k >> 1) << 7) + (j << 1) + (k & 1)] = Wn2[k * 64 + j];
  }
  __syncthreads();

  float* Aw = Al + wave * (16 * 128);
  const int totalWaves = gridDim.x * (blockDim.x >> 5);
  const int waveId = blockIdx.x * (blockDim.x >> 5) + wave;

  for (int tile = waveId; tile < Nn / 16; tile += totalWaves) {
    const int nb = tile << 4;
    // x (8 chunks/row) and u (8 chunks/row): per-lane async; agg needs math -> VALU path
    for (int idx = lane; idx < 256; idx += 32) {
      const int rr = idx >> 4, ch = idx & 15;
      const int n = nb + rr;
      if (ch < 8) async_b128(xf + (size_t)n * 32 + ch * 4, Aw + rr * 128 + ch * 4);
      else        async_b128(ubase + (size_t)(n % 384) * 32 + (ch - 8) * 4,
                             Aw + rr * 128 + 96 + (ch - 8) * 4);
    }
    for (int idx = lane; idx < 16 * 64; idx += 32) {
      const int rr = idx >> 6, kk = idx & 63;
      const int n = nb + rr;
      Aw[rr * 128 + 32 + kk] = sums[(size_t)n * 64 + kk] / fmaxf(cntf[n], 1.0f);
    }
    async_wait0();
    __syncthreads();

    v8f acc[4] = {v8f{}, v8f{}, v8f{}, v8f{}};
#pragma unroll
    for (int ks = 0; ks < 32; ++ks) {        // K = 128
      const int k0 = (ks << 2) + koff;
      const v2f a = ld2(Aw + m * 128 + k0);
      const float* wrow = Wl + ((k0 >> 1) << 7);
#pragma unroll
      for (int nt = 0; nt < 4; ++nt)
        acc[nt] = wmma_f32(a, ld2(wrow + ((nt * 16 + m) << 1)), acc[nt]);
    }
#pragma unroll
    for (int nt = 0; nt < 4; ++nt) {
      const float bias = bn2[nt * 16 + m];
#pragma unroll
      for (int r = 0; r < 8; ++r) {
        const int n = nb + r + hi * 8;
        const int g = n / Sn, s = n - g * Sn;
        const int bb = g / 24, tt = g - bb * 24;
        const int q = bb * Sn + s;                 // seq index b*S+s
        hxseq[((size_t)q * 24 + tt) * 64 + nt * 16 + m] = reluf(acc[nt][r] + bias);
      }
    }
    __syncthreads();
  }
}

// ---------------- kernel 5: encoder LSTM, persistent, weights in LDS ----------------
// Gate weights K-pair interleaved: W[(k>>1)*512 + 2*col + (k&1)], col = gate*64 + j
__global__ void enc_kernel(const float* __restrict__ hxseq,
                           const float* __restrict__ h0g, const float* __restrict__ c0g,
                           const float* __restrict__ Wih, const float* __restrict__ bih,
                           const float* __restrict__ Whh, const float* __restrict__ bhh,
                           float* __restrict__ h_enc, float* __restrict__ c_enc)
{
  extern __shared__ float smem[];
  float* WihL  = smem;                    // 32 k-pairs x 512
  float* WhhL  = WihL + 64 * 256;         // 32 k-pairs x 512
  float* biasL = WhhL + 64 * 256;         // [256] (bih+bhh)
  float* hbufs = biasL + 256;             // NWAVES * 2 * [16][64]

  const int tid = threadIdx.x, wave = tid >> 5, lane = tid & 31;
  const int m = lane & 15, hi = lane >> 4, koff = hi << 1;

  for (int i = tid; i < 64 * 256; i += blockDim.x) {
    const int k = i >> 8, j = i & 255;
    const int di = ((k >> 1) << 9) + (j << 1) + (k & 1);
    WihL[di] = Wih[i]; WhhL[di] = Whh[i];
  }
  for (int i = tid; i < 256; i += blockDim.x) biasL[i] = bih[i] + bhh[i];

  const int waveId = blockIdx.x * NWAVES_LSTM + wave;
  const int qb = waveId << 4;             // 16 sequences per wave
  float* hA = hbufs + wave * 2048;
  float* hB = hA + 1024;

  for (int i = lane; i < 1024; i += 32) hA[i] = h0g[(size_t)(qb + (i >> 6)) * 64 + (i & 63)];
  v8f c[4];
#pragma unroll
  for (int nt = 0; nt < 4; ++nt)
#pragma unroll
    for (int r = 0; r < 8; ++r)
      c[nt][r] = c0g[(size_t)(qb + r + hi * 8) * 64 + nt * 16 + m];
  __syncthreads();

  float* hcur = hA; float* hnxt = hB;
#pragma unroll 1
  for (int t = 0; t < 24; ++t) {
    const float* xrow = hxseq + ((size_t)(qb + m) * 24 + t) * 64;
    const float* hrow = hcur + m * 64;
    if (t < 23) __builtin_prefetch(xrow + 64, 0, 1);   // next timestep's input row
#pragma unroll
    for (int nt = 0; nt < 4; ++nt) {
      const int j = nt * 16 + m;
      v8f ai = {}, af = {}, ag = {}, ao = {};
#pragma unroll
      for (int ks = 0; ks < 16; ++ks) {     // x @ Wih, K=64 (4 independent gate chains)
        const int k0 = (ks << 2) + koff;
        const v2f a = ld2(xrow + k0);
        const float* wrow = WihL + ((k0 >> 1) << 9);
        ai = wmma_f32(a, ld2(wrow + (j << 1)), ai);
        af = wmma_f32(a, ld2(wrow + ((64 + j) << 1)), af);
        ag = wmma_f32(a, ld2(wrow + ((128 + j) << 1)), ag);
        ao = wmma_f32(a, ld2(wrow + ((192 + j) << 1)), ao);
      }
#pragma unroll
      for (int ks = 0; ks < 16; ++ks) {     // h @ Whh, K=64
        const int k0 = (ks << 2) + koff;
        const v2f a = ld2(hrow + k0);
        const float* wrow = WhhL + ((k0 >> 1) << 9);
        ai = wmma_f32(a, ld2(wrow + (j << 1)), ai);
        af = wmma_f32(a, ld2(wrow + ((64 + j) << 1)), af);
        ag = wmma_f32(a, ld2(wrow + ((128 + j) << 1)), ag);
        ao = wmma_f32(a, ld2(wrow + ((192 + j) << 1)), ao);
      }
      const float bi = biasL[j], bf = biasL[64 + j], bg = biasL[128 + j], bo = biasL[192 + j];
#pragma unroll
      for (int r = 0; r < 8; ++r) {
        const float iv = sigf(ai[r] + bi);
        const float fv = sigf(af[r] + bf);
        const float gv = tanhf(ag[r] + bg);
        const float ov = sigf(ao[r] + bo);
        const float cc = fv * c[nt][r] + iv * gv;
        c[nt][r] = cc;
        hnxt[(r + hi * 8) * 64 + j] = ov * tanhf(cc);   // D-layout -> row-major via LDS
      }
    }
    { float* tmp = hcur; hcur = hnxt; hnxt = tmp; }
    __syncthreads();
  }
  for (int i = lane; i < 1024; i += 32) h_enc[(size_t)(qb + (i >> 6)) * 64 + (i & 63)] = hcur[i];
#pragma unroll
  for (int nt = 0; nt < 4; ++nt)
#pragma unroll
    for (int r = 0; r < 8; ++r)
      c_enc[(size_t)(qb + r + hi * 8) * 64 + nt * 16 + m] = c[nt][r];
}

// ---------------- kernel 6: autoregressive decoder LSTM ----------------
__global__ void dec_kernel(const float* __restrict__ sta_aqi, const float* __restrict__ sta_for,
                           const float* __restrict__ h_enc, const float* __restrict__ c_enc,
                           const float* __restrict__ Wde, const float* __restrict__ bde,
                           const float* __restrict__ Wih, const float* __restrict__ bih,
                           const float* __restrict__ Whh, const float* __restrict__ bhh,
                           const float* __restrict__ Wlin, const float* __restrict__ blin,
                           float* __restrict__ out)
{
  extern __shared__ float smem[];
  float* WihL  = smem;                    // 10 k-pairs x 512 (K=20)
  float* WhhL  = WihL + 20 * 256;         // 32 k-pairs x 512
  float* biasL = WhhL + 64 * 256;         // [256]
  float* WlinL = biasL + 256;             // [64]
  float* WdeL  = WlinL + 64;              // [16]
  float* bdeL  = WdeL + 16;               // [16]
  float* hbufs = bdeL + 16;               // NWAVES * 2 * 1024
  float* inpL  = hbufs + NWAVES_LSTM * 2048;  // NWAVES * [16][20]
  float* aLs   = inpL + NWAVES_LSTM * 320;    // NWAVES * [16]

  const int tid = threadIdx.x, wave = tid >> 5, lane = tid & 31;
  const int m = lane & 15, hi = lane >> 4, koff = hi << 1;

  for (int i = tid; i < 20 * 256; i += blockDim.x) {
    const int k = i >> 8, j = i & 255;
    WihL[((k >> 1) << 9) + (j << 1) + (k & 1)] = Wih[i];
  }
  for (int i = tid; i < 64 * 256; i += blockDim.x) {
    const int k = i >> 8, j = i & 255;
    WhhL[((k >> 1) << 9) + (j << 1) + (k & 1)] = Whh[i];
  }
  for (int i = tid; i < 256; i += blockDim.x) biasL[i] = bih[i] + bhh[i];
  for (int i = tid; i < 64; i += blockDim.x) WlinL[i] = Wlin[i];
  if (tid < 16) { WdeL[tid] = Wde[tid]; bdeL[tid] = bde[tid]; }

  const int waveId = blockIdx.x * NWAVES_LSTM + wave;
  const int qb = waveId << 4;
  float* hA = hbufs + wave * 2048; float* hB = hA + 1024;
  float* inpw = inpL + wave * 320;
  float* aw = aLs + wave * 16;

  for (int i = lane; i < 1024; i += 32) hA[i] = h_enc[(size_t)(qb + (i >> 6)) * 64 + (i & 63)];
  v8f c[4];
#pragma unroll
  for (int nt = 0; nt < 4; ++nt)
#pragma unroll
    for (int r = 0; r < 8; ++r)
      c[nt][r] = c_enc[(size_t)(qb + r + hi * 8) * 64 + nt * 16 + m];
  if (lane < 16) aw[lane] = sta_aqi[(size_t)(qb + lane) * 24 + 23];  // last AQI hour
  __syncthreads();

  float* hcur = hA; float* hnxt = hB;
  const float bl = blin[0];
#pragma unroll 1
  for (int t = 0; t < Tn; ++t) {
    // build inp[16][20]: [0:16)=relu(a*Wde+bde), [16:20)=forecast; (qb+rr)%B == rr
    for (int idx = lane; idx < 320; idx += 32) {
      const int rr = idx / 20, kk = idx - rr * 20;
      float v;
      if (kk < 16) { const float a = aw[rr]; v = reluf(a * WdeL[kk] + bdeL[kk]); }
      else         v = sta_for[((size_t)rr * Tn + t) * 4 + (kk - 16)];
      inpw[idx] = v;
    }
    __syncthreads();

    const float* irow = inpw + m * 20;
    const float* hrow = hcur + m * 64;
#pragma unroll
    for (int nt = 0; nt < 4; ++nt) {
      const int j = nt * 16 + m;
      v8f ai = {}, af = {}, ag = {}, ao = {};
#pragma unroll
      for (int ks = 0; ks < 5; ++ks) {      // inp @ Wih, K=20
        const int k0 = (ks << 2) + koff;
        const v2f a = ld2(irow + k0);
        const float* wrow = WihL + ((k0 >> 1) << 9);
        ai = wmma_f32(a, ld2(wrow + (j << 1)), ai);
        af = wmma_f32(a, ld2(wrow + ((64 + j) << 1)), af);
        ag = wmma_f32(a, ld2(wrow + ((128 + j) << 1)), ag);
        ao = wmma_f32(a, ld2(wrow + ((192 + j) << 1)), ao);
      }
#pragma unroll
      for (int ks = 0; ks < 16; ++ks) {     // h @ Whh, K=64
        const int k0 = (ks << 2) + koff;
        const v2f a = ld2(hrow + k0);
        const float* wrow = WhhL + ((k0 >> 1) << 9);
        ai = wmma_f32(a, ld2(wrow + (j << 1)), ai);
        af = wmma_f32(a, ld2(wrow + ((64 + j) << 1)), af);
        ag = wmma_f32(a, ld2(wrow + ((128 + j) << 1)), ag);
        ao = wmma_f32(a, ld2(wrow + ((192 + j) << 1)), ao);
      }
      const float bi = biasL[j], bf = biasL[64 + j], bg = biasL[128 + j], bo = biasL[192 + j];
#pragma unroll
      for (int r = 0; r < 8; ++r) {
        const float iv = sigf(ai[r] + bi);
        const float fv = sigf(af[r] + bf);
        const float gv = tanhf(ag[r] + bg);
        const float ov = sigf(ao[r] + bo);
        const float cc = fv * c[nt][r] + iv * gv;
        c[nt][r] = cc;
        hnxt[(r + hi * 8) * 64 + j] = ov * tanhf(cc);
      }
    }
    { float* tmp = hcur; hcur = hnxt; hnxt = tmp; }  // hcur = new h
    __syncthreads();

    if (lane < 16) {                        // y = relu(h @ W_lin + b_lin)
      float sacc = bl;
#pragma unroll
      for (int k = 0; k < 64; ++k) sacc += hcur[lane * 64 + k] * WlinL[k];
      sacc = reluf(sacc);
      out[(size_t)(qb + lane) * Tn + t] = sacc;      // [B,S,T] with q=b*S+s
      aw[lane] = sacc;
    }
    __syncthreads();
  }
}

// ---------------- host launch ----------------
extern "C" void kernel_launch(void* const* d_in, const int* in_sizes, int n_in,
                              void* d_out, int out_size, void* d_ws, size_t ws_size,
                              hipStream_t stream)
{
  (void)in_sizes; (void)n_in; (void)out_size; (void)ws_size;
  const float* sta_aqi  = (const float*)d_in[0];
  const int*   sta_conn = (const int*)  d_in[1];
  const float* sta_poi  = (const float*)d_in[2];
  const float* sta_w    = (const float*)d_in[3];
  const float* sta_wea  = (const float*)d_in[4];
  const float* sta_for  = (const float*)d_in[5];
  const float* city_u   = (const float*)d_in[6];
  const float* h0       = (const float*)d_in[7];
  const float* c0       = (const float*)d_in[8];
  const float* W_aqi    = (const float*)d_in[9];
  const float* b_aqi    = (const float*)d_in[10];
  const float* W_poi    = (const float*)d_in[11];
  const float* b_poi    = (const float*)d_in[12];
  const float* W_city   = (const float*)d_in[13];
  const float* b_city   = (const float*)d_in[14];
  const float* W_wea    = (const float*)d_in[15];
  const float* b_wea    = (const float*)d_in[16];
  const float* W_n1     = (const float*)d_in[17];
  const float* b_n1     = (const float*)d_in[18];
  const float* W_n2     = (const float*)d_in[19];
  const float* b_n2     = (const float*)d_in[20];
  const float* enc_Wih  = (const float*)d_in[21];
  const float* enc_bih  = (const float*)d_in[22];
  const float* enc_Whh  = (const float*)d_in[23];
  const float* enc_bhh  = (const float*)d_in[24];
  const float* W_dec_em = (const float*)d_in[25];
  const float* b_dec_em = (const float*)d_in[26];
  const float* dec_Wih  = (const float*)d_in[27];
  const float* dec_bih  = (const float*)d_in[28];
  const float* dec_Whh  = (const float*)d_in[29];
  const float* dec_bhh  = (const float*)d_in[30];
  const float* W_lin    = (const float*)d_in[31];
  const float* b_lin    = (const float*)d_in[32];

  float* ws    = (float*)d_ws;
  float* xf    = ws;                                    // N*32
  float* ubase = xf + (size_t)Nn * 32;                  // 384*32
  float* sums  = ubase + 384 * 32;                      // N*64
  float* cntf  = sums + (size_t)Nn * 64;                // N
  float* hxseq = cntf + Nn;                             // BS*24*64
  float* h_enc = hxseq + (size_t)BSn * 24 * 64;         // BS*64
  float* c_enc = h_enc + (size_t)BSn * 64;              // BS*64

  // zero the scatter accumulators (sums + cnt are contiguous: N*65 floats)
  hipMemsetAsync(sums, 0, (size_t)Nn * 65 * sizeof(float), stream);

  embed_kernel<<<(Nn + 255) / 256, 256, 0, stream>>>(sta_aqi, sta_poi, W_aqi, b_aqi, W_poi, b_poi, xf);
  ubase_kernel<<<(384 * 32 + 255) / 256, 256, 0, stream>>>(city_u, sta_wea, W_city, b_city, W_wea, b_wea, ubase);

  // 192 blocks * 8 waves = 1536 waves; 49152 edge tiles -> 32 tiles/wave (uniform barriers)
  const size_t msgSh = (size_t)(68 * 64 + 8 * 16 * 68 + 8 * 16) * sizeof(float);
  gnn_msg_kernel<<<192, 256, msgSh, stream>>>(xf, sta_conn, sta_w, W_n1, b_n1, sums, cntf);

  // 96 blocks * 8 waves = 768 waves; 6144 node tiles -> 8 tiles/wave (uniform)
  const size_t nodeSh = (size_t)(128 * 64 + 8 * 16 * 128) * sizeof(float);
  gnn_node_kernel<<<96, 256, nodeSh, stream>>>(xf, sums, cntf, ubase, W_n2, b_n2, hxseq);

  // 64 blocks * 4 waves = 256 waves; 4096 seqs -> one 16-seq tile per wave, persistent
  const size_t encSh = (size_t)(64 * 256 * 2 + 256 + NWAVES_LSTM * 2048) * sizeof(float);
  enc_kernel<<<64, 128, encSh, stream>>>(hxseq, h0, c0, enc_Wih, enc_bih, enc_Whh, enc_bhh, h_enc, c_enc);

  const size_t decSh = (size_t)(20 * 256 + 64 * 256 + 256 + 64 + 16 + 16 +
                                NWAVES_LSTM * 2048 + NWAVES_LSTM * 320 + NWAVES_LSTM * 16) * sizeof(float);
  dec_kernel<<<64, 128, decSh, stream>>>(sta_aqi, sta_for, h_enc, c_enc, W_dec_em, b_dec_em,
                                         dec_Wih, dec_bih, dec_Whh, dec_bhh, W_lin, b_lin,
                                         (float*)d_out);
}